// Net_64725157151033
// MI455X (gfx1250) — compile-verified
//
#include <hip/hip_runtime.h>
#include <hip/hip_bf16.h>

typedef __attribute__((ext_vector_type(2))) float v2f;
typedef __attribute__((ext_vector_type(8))) float v8f;

#define NODES 100000
#define FDIM 64

// ---------------------------------------------------------------------------
// Degree: one f32 atomic per edge (computed once, reused by both layers).
// ---------------------------------------------------------------------------
__global__ void deg_kernel(const int* __restrict__ eidx, int E,
                           float* __restrict__ deg) {
    int e = blockIdx.x * blockDim.x + threadIdx.x;
    if (e < E) {
        int dst = eidx[E + e];
        atomicAdd(&deg[dst], 1.0f);
    }
}

// ---------------------------------------------------------------------------
// Convert degree -> reciprocal once (keeps IEEE divides out of GEMM waves).
// ---------------------------------------------------------------------------
__global__ void rdeg_kernel(float* __restrict__ deg, int N) {
    int i = blockIdx.x * blockDim.x + threadIdx.x;
    if (i < N) deg[i] = 1.0f / fmaxf(deg[i], 1.0f);
}

// ---------------------------------------------------------------------------
// Scatter-add of 64-dim features: 16 lanes per edge, float4 gather per lane
// (coalesced 256B row read, L2-resident), 4 f32 atomics per lane.
// ---------------------------------------------------------------------------
__global__ void scatter_kernel(const float* __restrict__ feat,
                               const int* __restrict__ eidx, int E,
                               float* __restrict__ agg) {
    long long t = (long long)blockIdx.x * blockDim.x + threadIdx.x;
    int e  = (int)(t >> 4);
    int c4 = ((int)t & 15) * 4;
    if (e >= E) return;
    int src = eidx[e];
    int dst = eidx[E + e];
    const float4 v = *(const float4*)(feat + (long long)src * FDIM + c4);
    float* base = agg + (long long)dst * FDIM + c4;
    atomicAdd(base + 0, v.x);
    atomicAdd(base + 1, v.y);
    atomicAdd(base + 2, v.z);
    atomicAdd(base + 3, v.w);
}

// ---------------------------------------------------------------------------
// Fused SAGE transform for one layer:
//   out[m, :] = relu?( agg[m,:]*rdeg[m] @ Wn + feat[m,:] @ Ws + bias )
// One wave computes a 16 x NCOLS tile with V_WMMA_F32_16X16X4_F32 (full f32,
// matches reference precision; FLOPs are trivial, this phase is bandwidth).
//
// Weights are staged once per block into LDS, pair-interleaved so each B
// operand (W[k][n], W[k+1][n]) is a single conflict-free ds_load_b64.
//
// f32 WMMA operand layout (wave32):
//   A 16x4 : lane L -> row M = L%16, holds K = (L/16)*2 + {0,1}  (v2f)
//   B 4x16 : lane L -> col N = L%16, holds K = (L/16)*2 + {0,1}  (v2f)
//   C 16x16: vgpr i, lane L -> M = i + (L/16)*8, N = L%16        (v8f)
// ---------------------------------------------------------------------------
template <int NCOLS, bool RELU>
__global__ void sage_gemm_kernel(const float* __restrict__ agg,
                                 const float* __restrict__ rdeg,
                                 const float* __restrict__ feat,
                                 const float* __restrict__ Wn,   // [64 x NCOLS]
                                 const float* __restrict__ Ws,   // [64 x NCOLS]
                                 const float* __restrict__ bias, // [NCOLS]
                                 float* __restrict__ out,        // [N x NCOLS]
                                 int N) {
    constexpr int NT    = NCOLS / 16;
    constexpr int PAIRS = (FDIM / 2) * NCOLS;   // k-pairs * cols

    // LDS: pair-interleaved weights. ldsW*[p*NCOLS + n] = {W[2p][n], W[2p+1][n]}
    __shared__ v2f ldsWn[PAIRS];
    __shared__ v2f ldsWs[PAIRS];

    // ---- cooperative weight staging (all 256 threads) ----
    for (int pi = threadIdx.x; pi < PAIRS; pi += blockDim.x) {
        const int p = pi / NCOLS;
        const int n = pi - p * NCOLS;
        v2f wn, wsv;
        wn.x  = Wn[(2 * p)     * NCOLS + n];
        wn.y  = Wn[(2 * p + 1) * NCOLS + n];
        wsv.x = Ws[(2 * p)     * NCOLS + n];
        wsv.y = Ws[(2 * p + 1) * NCOLS + n];
        ldsWn[pi] = wn;
        ldsWs[pi] = wsv;
    }
    __syncthreads();

    const int wid  = ((blockIdx.x * blockDim.x + threadIdx.x) >> 5);
    const int lane = threadIdx.x & 31;
    const int row0 = wid * 16;
    if (row0 >= N) return;  // uniform per wave (after the barrier)

    const int mrow  = row0 + (lane & 15);
    const int khalf = (lane >> 4) * 2;  // 0 or 2
    const int ncol  = lane & 15;

    const float rd = rdeg[mrow];        // precomputed 1/max(deg,1)

    const float* __restrict__ arow = agg  + (long long)mrow * FDIM;
    const float* __restrict__ xrow = feat + (long long)mrow * FDIM;

    v8f acc[NT] = {};

    #pragma unroll
    for (int k0 = 0; k0 < FDIM; k0 += 4) {
        const int ka = k0 + khalf;          // even
        const int p  = ka >> 1;             // k-pair index
        // A operands: one b64 load each (ka even -> 8B aligned)
        v2f a_m = *(const v2f*)(arow + ka);
        a_m *= rd;                          // v_pk_mul_f32
        v2f a_x = *(const v2f*)(xrow + ka);
        #pragma unroll
        for (int t = 0; t < NT; ++t) {
            const int n = t * 16 + ncol;
            v2f b_n = ldsWn[p * NCOLS + n]; // ds_load_b64, conflict-free
            v2f b_s = ldsWs[p * NCOLS + n];
            acc[t] = __builtin_amdgcn_wmma_f32_16x16x4_f32(
                false, a_m, false, b_n, (short)0, acc[t], false, false);
            acc[t] = __builtin_amdgcn_wmma_f32_16x16x4_f32(
                false, a_x, false, b_s, (short)0, acc[t], false, false);
        }
    }

    // Bias + (optional) ReLU + store per the C/D striping.
    const int mbase = row0 + (lane >> 4) * 8;
    #pragma unroll
    for (int t = 0; t < NT; ++t) {
        const int n    = t * 16 + ncol;
        const float bv = bias[n];
        #pragma unroll
        for (int i = 0; i < 8; ++i) {
            float v = acc[t][i] + bv;
            if (RELU) v = fmaxf(v, 0.0f);
            out[(long long)(mbase + i) * NCOLS + n] = v;
        }
    }
}

// ---------------------------------------------------------------------------
// Host launcher (graph-capture safe: only async ops on `stream`).
// ---------------------------------------------------------------------------
extern "C" void kernel_launch(void* const* d_in, const int* in_sizes, int n_in,
                              void* d_out, int out_size, void* d_ws, size_t ws_size,
                              hipStream_t stream) {
    const float* x   = (const float*)d_in[0];
    const int*   eix = (const int*)d_in[1];     // [2, E] int32 (JAX default x64-off)
    const float* W1n = (const float*)d_in[2];
    const float* W1s = (const float*)d_in[3];
    const float* b1  = (const float*)d_in[4];
    const float* W2n = (const float*)d_in[5];
    const float* W2s = (const float*)d_in[6];
    const float* b2  = (const float*)d_in[7];
    float* out = (float*)d_out;

    const int N = NODES;
    const int E = in_sizes[1] / 2;

    // Workspace layout: rdeg [N] | agg [N*64] | h [N*64]
    char* ws = (char*)d_ws;
    float* deg = (float*)ws;
    float* agg = (float*)(ws + (size_t)N * sizeof(float));
    float* h   = (float*)(ws + (size_t)N * sizeof(float) +
                               (size_t)N * FDIM * sizeof(float));

    const int BLK = 256;
    const int deg_blocks  = (E + BLK - 1) / BLK;
    const int rd_blocks   = (N + BLK - 1) / BLK;
    const long long sc_th = (long long)E * 16;
    const int sc_blocks   = (int)((sc_th + BLK - 1) / BLK);
    const int tiles       = (N + 15) / 16;          // 6250
    const int gemm_blocks = (tiles + 7) / 8;        // 8 waves / block

    // ---- degree -> reciprocal (once) ----
    hipMemsetAsync(deg, 0, (size_t)N * sizeof(float), stream);
    deg_kernel<<<deg_blocks, BLK, 0, stream>>>(eix, E, deg);
    rdeg_kernel<<<rd_blocks, BLK, 0, stream>>>(deg, N);

    // ---- layer 1 ----
    hipMemsetAsync(agg, 0, (size_t)N * FDIM * sizeof(float), stream);
    scatter_kernel<<<sc_blocks, BLK, 0, stream>>>(x, eix, E, agg);
    sage_gemm_kernel<64, true><<<gemm_blocks, BLK, 0, stream>>>(
        agg, deg, x, W1n, W1s, b1, h, N);

    // ---- layer 2 ----
    hipMemsetAsync(agg, 0, (size_t)N * FDIM * sizeof(float), stream);
    scatter_kernel<<<sc_blocks, BLK, 0, stream>>>(h, eix, E, agg);
    sage_gemm_kernel<32, false><<<gemm_blocks, BLK, 0, stream>>>(
        agg, deg, h, W2n, W2s, b2, out, N);
}